// EdgeNet_64570538328684
// MI455X (gfx1250) — compile-verified
//
#include <hip/hip_runtime.h>
#include <math.h>

#define N_NODES 4096
#define N_EDGES 16384

typedef float v2f __attribute__((ext_vector_type(2)));
typedef float v8f __attribute__((ext_vector_type(8)));

// ---------------------------------------------------------------------------
// Stage 1: B[e,0:4] = (Ro^T X)[e], B[e,4:8] = (Ri^T X)[e]   via f32 WMMA 16x16x4
// One wave handles a 16-edge tile over a 1024-node K-chunk (split-K x4).
// A operand: lanes 0-15 -> Ro rows (K=0,1), lanes 16-31 -> Ri rows (K=2,3).
// B operand: block-diagonal X so cols 0-3 accumulate Ro^T X, cols 4-7 Ri^T X.
// Inner loop is branch-free: unconditional LDS loads masked by a 0/1 scale,
// keeping EXEC all-ones (required by WMMA) and the load pipeline clean.
// ---------------------------------------------------------------------------
__global__ __launch_bounds__(256) void edgenet_gemm_wmma(
    const float* __restrict__ X,
    const float* __restrict__ Ri,
    const float* __restrict__ Ro,
    float* __restrict__ Bout) {
  __shared__ float sX[N_NODES * 4];  // 64 KB: X staged once per block

  for (int i = threadIdx.x; i < N_NODES * 4; i += blockDim.x) sX[i] = X[i];
  __syncthreads();

  const int lane = threadIdx.x & 31;
  const int wave = threadIdx.x >> 5;              // 8 waves per block
  const int tile = blockIdx.x * 2 + (wave >> 2);  // 2 edge-tiles per block
  const int kchunk = wave & 3;                    // split-K: 4 chunks of 1024
  const int e0 = tile * 16;
  const int m = lane & 15;                        // edge within tile / B column
  const bool hi = lane >= 16;                     // hi half: Ri (K=2,3)

  const float* __restrict__ Abase = (hi ? Ri : Ro) + (size_t)e0 + m;

  // B-operand lane setup: lo lanes feed cols 0-3 (Ro*X), hi lanes cols 4-7
  // (Ri*X); all other columns must contribute zero -> 0/1 scale, no branches.
  const bool useB = hi ? (m >= 4 && m < 8) : (m < 4);
  const float bscale = useB ? 1.f : 0.f;
  const int xcol = m & 3;

  v8f acc0 = {0.f, 0.f, 0.f, 0.f, 0.f, 0.f, 0.f, 0.f};
  v8f acc1 = acc0;

  const int n0 = kchunk * (N_NODES / 4);
  const int n1 = n0 + (N_NODES / 4);
#pragma unroll 8
  for (int n = n0; n < n1; n += 2) {
    v2f a;
    a.x = Abase[(size_t)n * N_EDGES];        // K = 0 (lo) / K = 2 (hi)
    a.y = Abase[(size_t)(n + 1) * N_EDGES];  // K = 1 (lo) / K = 3 (hi)
    v2f b;
    b.x = sX[n * 4 + xcol] * bscale;
    b.y = sX[n * 4 + 4 + xcol] * bscale;
    if ((n >> 1) & 1) {
      acc1 = __builtin_amdgcn_wmma_f32_16x16x4_f32(
          false, a, false, b, (short)0, acc1, false, false);
    } else {
      acc0 = __builtin_amdgcn_wmma_f32_16x16x4_f32(
          false, a, false, b, (short)0, acc0, false, false);
    }
  }

  // Split-K reduction through LDS (reuse sX after all waves finish the loop).
  __syncthreads();
  float* red = sX;  // 256 threads * 8 floats = 8 KB
#pragma unroll
  for (int r = 0; r < 8; ++r) red[threadIdx.x * 8 + r] = acc0[r] + acc1[r];
  __syncthreads();

  if ((wave & 3) == 0 && m < 8) {
#pragma unroll
    for (int r = 0; r < 8; ++r) {
      float v = red[(threadIdx.x + 0) * 8 + r] + red[(threadIdx.x + 32) * 8 + r] +
                red[(threadIdx.x + 64) * 8 + r] + red[(threadIdx.x + 96) * 8 + r];
      const int row = (hi ? 8 : 0) + r;  // C/D layout: VGPR r = M=r (lo), M=r+8 (hi)
      Bout[(size_t)(e0 + row) * 8 + m] = v;
    }
  }
}

// ---------------------------------------------------------------------------
// Stage 2: 8-qubit circuit per edge; one wave32 per edge.
// Amplitude index a[7:0]: bits 0-2 = in-lane index j, bits 3-7 = lane id.
// ---------------------------------------------------------------------------
struct U2 {
  float r00, i00, r01, i01, r10, i10, r11, i11;
};

__device__ inline U2 make_rot(const float* __restrict__ th, int i) {
  // PennyLane Rot = RZ(omega) RY(theta) RZ(phi)
  const float phi = th[i], t = th[i + 1], om = th[i + 2];
  float st, ct, sa, ca, sd, cd;
  __sincosf(0.5f * t, &st, &ct);
  __sincosf(0.5f * (phi + om), &sa, &ca);
  __sincosf(0.5f * (phi - om), &sd, &cd);
  U2 u;
  u.r00 = ct * ca;  u.i00 = -ct * sa;
  u.r11 = ct * ca;  u.i11 =  ct * sa;
  u.r01 = -st * cd; u.i01 = -st * sd;
  u.r10 =  st * cd; u.i10 = -st * sd;
  return u;
}

template <int M>
__device__ inline void rot_local(float (&ar)[8], float (&ai)[8], const U2& u) {
#pragma unroll
  for (int j = 0; j < 8; ++j) {
    if ((j & M) == 0) {
      const int k = j | M;
      const float x0r = ar[j], x0i = ai[j], x1r = ar[k], x1i = ai[k];
      ar[j] = u.r00 * x0r - u.i00 * x0i + u.r01 * x1r - u.i01 * x1i;
      ai[j] = u.r00 * x0i + u.i00 * x0r + u.r01 * x1i + u.i01 * x1r;
      ar[k] = u.r10 * x0r - u.i10 * x0i + u.r11 * x1r - u.i11 * x1i;
      ai[k] = u.r10 * x0i + u.i10 * x0r + u.r11 * x1i + u.i11 * x1r;
    }
  }
}

template <int L>
__device__ inline void rot_lane(float (&ar)[8], float (&ai)[8], const U2& u, int lane) {
  const bool hb = (lane & L) != 0;
  const float g0r = hb ? u.r10 : u.r00, g0i = hb ? u.i10 : u.i00;
  const float g1r = hb ? u.r11 : u.r01, g1i = hb ? u.i11 : u.i01;
#pragma unroll
  for (int j = 0; j < 8; ++j) {
    const float pr = __shfl_xor(ar[j], L, 32);
    const float pi = __shfl_xor(ai[j], L, 32);
    const float x0r = hb ? pr : ar[j], x0i = hb ? pi : ai[j];
    const float x1r = hb ? ar[j] : pr, x1i = hb ? ai[j] : pi;
    ar[j] = g0r * x0r - g0i * x0i + g1r * x1r - g1i * x1i;
    ai[j] = g0r * x0i + g0i * x0r + g1r * x1i + g1i * x1r;
  }
}

template <int MC, int MT>
__device__ inline void cnot_local_local(float (&ar)[8], float (&ai)[8]) {
#pragma unroll
  for (int j = 0; j < 8; ++j) {
    if ((j & MC) && !(j & MT)) {
      const int k = j | MT;
      float t = ar[j]; ar[j] = ar[k]; ar[k] = t;
      t = ai[j]; ai[j] = ai[k]; ai[k] = t;
    }
  }
}

template <int LC, int MT>
__device__ inline void cnot_lane_local(float (&ar)[8], float (&ai)[8], int lane) {
  const bool cc = (lane & LC) != 0;
#pragma unroll
  for (int j = 0; j < 8; ++j) {
    if (!(j & MT)) {
      const int k = j | MT;
      const float njr = cc ? ar[k] : ar[j], nkr = cc ? ar[j] : ar[k];
      const float nji = cc ? ai[k] : ai[j], nki = cc ? ai[j] : ai[k];
      ar[j] = njr; ar[k] = nkr; ai[j] = nji; ai[k] = nki;
    }
  }
}

template <int MC, int LT>
__device__ inline void cnot_local_lane(float (&ar)[8], float (&ai)[8]) {
#pragma unroll
  for (int j = 0; j < 8; ++j) {
    if (j & MC) {
      ar[j] = __shfl_xor(ar[j], LT, 32);
      ai[j] = __shfl_xor(ai[j], LT, 32);
    }
  }
}

template <int LC, int LT>
__device__ inline void cnot_lane_lane(float (&ar)[8], float (&ai)[8], int lane) {
  const bool cc = (lane & LC) != 0;
#pragma unroll
  for (int j = 0; j < 8; ++j) {
    const float tr = __shfl_xor(ar[j], LT, 32);
    const float ti = __shfl_xor(ai[j], LT, 32);
    ar[j] = cc ? tr : ar[j];
    ai[j] = cc ? ti : ai[j];
  }
}

__global__ __launch_bounds__(256) void edgenet_circuit(
    const float* __restrict__ Bmat,
    const float* __restrict__ theta,
    float* __restrict__ out) {
  const int lane = threadIdx.x & 31;
  const int wave = threadIdx.x >> 5;
  const int e = blockIdx.x * 8 + wave;

  // Gather RY angles: lane q (q<8) holds B[e,q]; broadcast to all lanes.
  const float bq_lane = Bmat[(size_t)e * 8 + (lane & 7)];
  float c[8], s[8];
#pragma unroll
  for (int q = 0; q < 8; ++q) {
    const float bq = __shfl(bq_lane, q, 32);
    __sincosf(0.5f * bq, &s[q], &c[q]);
  }

  // Product state: amp(a) = prod_q (bit_q ? sin : cos). a = (lane<<3)|j.
  float ar[8], ai[8];
#pragma unroll
  for (int j = 0; j < 8; ++j) {
    const int a = (lane << 3) | j;
    float v = 1.f;
#pragma unroll
    for (int q = 0; q < 8; ++q) v *= ((a >> q) & 1) ? s[q] : c[q];
    ar[j] = v;
    ai[j] = 0.f;
  }

  // Gate sequence. Local masks: w0->1, w1->2, w2->4. Lane masks: w3->1,
  // w4->2, w5->4, w6->8, w7->16.
  { U2 u = make_rot(theta, 0);  rot_local<1>(ar, ai, u); }        // Rot w0
  { U2 u = make_rot(theta, 3);  rot_local<2>(ar, ai, u); }        // Rot w1
  cnot_local_local<1, 2>(ar, ai);                                 // CNOT(0,1)
  { U2 u = make_rot(theta, 6);  rot_local<4>(ar, ai, u); }        // Rot w2
  { U2 u = make_rot(theta, 9);  rot_lane<1>(ar, ai, u, lane); }   // Rot w3
  cnot_lane_local<1, 4>(ar, ai, lane);                            // CNOT(3,2)
  { U2 u = make_rot(theta, 12); rot_lane<2>(ar, ai, u, lane); }   // Rot w4
  { U2 u = make_rot(theta, 15); rot_lane<4>(ar, ai, u, lane); }   // Rot w5
  cnot_lane_lane<2, 4>(ar, ai, lane);                             // CNOT(4,5)
  { U2 u = make_rot(theta, 18); rot_lane<8>(ar, ai, u, lane); }   // Rot w6
  { U2 u = make_rot(theta, 21); rot_lane<16>(ar, ai, u, lane); }  // Rot w7
  cnot_lane_lane<16, 8>(ar, ai, lane);                            // CNOT(7,6)
  { U2 u = make_rot(theta, 24); rot_local<2>(ar, ai, u); }        // Rot w1
  { U2 u = make_rot(theta, 27); rot_local<4>(ar, ai, u); }        // Rot w2
  cnot_local_local<2, 4>(ar, ai);                                 // CNOT(1,2)
  { U2 u = make_rot(theta, 30); rot_lane<4>(ar, ai, u, lane); }   // Rot w5
  { U2 u = make_rot(theta, 33); rot_lane<8>(ar, ai, u, lane); }   // Rot w6
  cnot_lane_lane<8, 4>(ar, ai, lane);                             // CNOT(6,5)
  { U2 u = make_rot(theta, 36); rot_local<4>(ar, ai, u); }        // Rot w2
  { U2 u = make_rot(theta, 39); rot_lane<4>(ar, ai, u, lane); }   // Rot w5
  cnot_local_lane<4, 4>(ar, ai);                                  // CNOT(2,5)
  { U2 u = make_rot(theta, 42); rot_lane<4>(ar, ai, u, lane); }   // Rot w5

  // <Z> on wire 5 == lane bit 2.
  float p = 0.f;
#pragma unroll
  for (int j = 0; j < 8; ++j) p += ar[j] * ar[j] + ai[j] * ai[j];
  p = (lane & 4) ? -p : p;
#pragma unroll
  for (int off = 16; off > 0; off >>= 1) p += __shfl_xor(p, off, 32);
  if (lane == 0) out[e] = 0.5f * (1.f - p);
}

extern "C" void kernel_launch(void* const* d_in, const int* in_sizes, int n_in,
                              void* d_out, int out_size, void* d_ws, size_t ws_size,
                              hipStream_t stream) {
  (void)in_sizes; (void)n_in; (void)out_size; (void)ws_size;
  const float* X = (const float*)d_in[0];
  const float* Ri = (const float*)d_in[1];
  const float* Ro = (const float*)d_in[2];
  const float* theta = (const float*)d_in[3];
  float* out = (float*)d_out;
  float* Bmat = (float*)d_ws;  // 16384 * 8 floats = 512 KB scratch

  // 512 blocks * 8 waves = 4096 waves: 1024 edge-tiles x 4 K-chunks.
  edgenet_gemm_wmma<<<512, 256, 0, stream>>>(X, Ri, Ro, Bmat);
  // 2048 blocks * 8 waves = 16384 waves: one per edge.
  edgenet_circuit<<<2048, 256, 0, stream>>>(Bmat, theta, out);
}